// SAGE_3layer_17205638988374
// MI455X (gfx1250) — compile-verified
//
#include <hip/hip_runtime.h>

// ---------------------------------------------------------------------------
// GraphSAGE 3-layer forward for MI455X (gfx1250, wave32).
//
// Roofline: feature tables (25.6 MB) are L2-resident (192 MB L2) -> the
// gather/scatter phase is L2-atomic bound, not HBM bound. Total GEMM work is
// only ~10 GFLOP, so fp32 WMMA (V_WMMA_F32_16X16X4_F32) is used to keep
// reference-exact fp32 numerics while still running on the matrix pipe.
// Weight tiles are staged into LDS with CDNA5 async global->LDS copies
// (GLOBAL_LOAD_ASYNC_TO_LDS_B128, ASYNCcnt) instead of load/ds_store pairs.
// ---------------------------------------------------------------------------

typedef __attribute__((ext_vector_type(2))) float v2f;
typedef __attribute__((ext_vector_type(8))) float v8f;

#define NNODES 50000
#define NEDGES 600000
#define DF     128     // D_IN == D_HID
#define DOUT   16
#define NGRAPH 64
#define LWPAD  36      // LDS row stride (floats): 144B -> 16B aligned, banks ok

// ---------------------------------------------------------------- utilities
__global__ void k_zero(float* __restrict__ p, long long n) {
  long long i = (long long)blockIdx.x * blockDim.x + threadIdx.x;
  long long stride = (long long)gridDim.x * blockDim.x;
  for (; i < n; i += stride) p[i] = 0.0f;
}

__global__ void k_degree(const int* __restrict__ dst, float* __restrict__ deg, int E) {
  int e = blockIdx.x * blockDim.x + threadIdx.x;
  if (e < E) atomicAdd(&deg[dst[e]], 1.0f);
}

__global__ void k_inv_degree(float* __restrict__ deg, int n) {
  int i = blockIdx.x * blockDim.x + threadIdx.x;
  if (i < n) deg[i] = 1.0f / fmaxf(deg[i], 1.0f);
}

// One wave per edge: 32 lanes x float4 = one 128-float feature row.
__global__ void k_scatter(const float* __restrict__ hin, const int* __restrict__ src,
                          const int* __restrict__ dst, float* __restrict__ agg, int E) {
  int lane = threadIdx.x & 31;
  int e = blockIdx.x * (blockDim.x >> 5) + (threadIdx.x >> 5);
  if (e >= E) return;
  int s = src[e];
  int d = dst[e];
  const float4 v = *(const float4*)(hin + (size_t)s * DF + lane * 4);
  float* p = agg + (size_t)d * DF + lane * 4;
  atomicAdd(p + 0, v.x);
  atomicAdd(p + 1, v.y);
  atomicAdd(p + 2, v.z);
  atomicAdd(p + 3, v.w);
}

// ---------------------------------------------------------------- SAGE GEMM
// out[m, n] = relu( sum_k  A[m, k] * Wcat[n, k]  + bias[n] )
//   A[m, 0:128]   = agg[m, :] * inv_deg[m]   (mean aggregation)
//   A[m, 128:256] = hin[m, :]                (self term)
//   Wcat[n, :]    = [ w_l[n, :] | w_r[n, :] ]
// Block: 256 threads = 8 waves. Wave w owns rows m0 = blk*128 + w*16 .. +15
// and all 8 N-tiles (8 x v8f accumulators). K consumed in 32-wide weight
// chunks staged via async global->LDS DMA; WMMA f32 16x16x4 per step.
__global__ __launch_bounds__(256) void k_sage_gemm(
    const float* __restrict__ agg, const float* __restrict__ ideg,
    const float* __restrict__ hin,
    const float* __restrict__ wl, const float* __restrict__ wr,
    const float* __restrict__ bias, float* __restrict__ hout, int nrows) {
  __shared__ float lw[DF * LWPAD];  // 128 rows x 32-wide K chunk (padded rows)

  const int lane = threadIdx.x & 31;
  const int wv   = threadIdx.x >> 5;              // wave 0..7
  const int m0   = blockIdx.x * 128 + wv * 16;    // wave's 16-row base
  const int mrow = m0 + (lane & 15);
  const int mclamp = (mrow < nrows) ? mrow : (nrows - 1);  // keep EXEC all-ones
  const int khalf = (lane >> 4) << 1;             // lanes 0-15 -> K+0/1, 16-31 -> K+2/3

  const float invd = ideg[mclamp];
  const float* __restrict__ arow_agg = agg + (size_t)mclamp * DF;
  const float* __restrict__ arow_hin = hin + (size_t)mclamp * DF;

  // Generic LDS pointer: low 32 bits == wave-relative LDS byte offset.
  const unsigned lw_base = (unsigned)(uintptr_t)&lw[0];

  const v8f zero8 = {0.f, 0.f, 0.f, 0.f, 0.f, 0.f, 0.f, 0.f};
  v8f acc[8];
#pragma unroll
  for (int nt = 0; nt < 8; ++nt) acc[nt] = zero8;

  for (int kc = 0; kc < 8; ++kc) {               // 8 chunks of 32 over K=256
    const float* __restrict__ wsrc = (kc < 4) ? wl : wr;
    const int kbase = (kc & 3) * 32;

    __syncthreads();  // all waves done reading the previous chunk

    // Async DMA the 128x32 weight chunk into LDS: 1024 float4 chunks,
    // 4 per thread, tracked by ASYNCcnt (no VGPR round-trip).
    {
      const float* gsrcbase = wsrc + kbase;
      int c = threadIdx.x;
#pragma unroll
      for (int j = 0; j < 4; ++j, c += 256) {
        const int n  = c >> 3;          // weight row (output feature)
        const int k4 = (c & 7) << 2;    // float4 offset inside the 32-wide chunk
        const unsigned ldsoff = lw_base + (unsigned)(n * LWPAD + k4) * 4u;
        const float* gp = gsrcbase + (size_t)n * DF + k4;
        asm volatile("global_load_async_to_lds_b128 %0, %1, off"
                     :: "v"(ldsoff), "v"(gp)
                     : "memory");
      }
      asm volatile("s_wait_asynccnt 0x0" ::: "memory");
    }
    __syncthreads();  // chunk visible to every wave

    const float* __restrict__ asrc = (kc < 4) ? arow_agg : arow_hin;
    const float sc = (kc < 4) ? invd : 1.0f;     // fold mean-scaling into A load

#pragma unroll
    for (int ks = 0; ks < 32; ks += 4) {
      const int kk = kbase + ks + khalf;
      v2f a;
      a.x = asrc[kk]     * sc;                   // A frag: 16x4 f32 ISA layout
      a.y = asrc[kk + 1] * sc;
#pragma unroll
      for (int nt = 0; nt < 8; ++nt) {
        const int nl = nt * 16 + (lane & 15);
        const int dk = ks + khalf;
        v2f b;                                   // B frag: B[k][n] = Wcat[n][k]
        b.x = lw[nl * LWPAD + dk];
        b.y = lw[nl * LWPAD + dk + 1];
        acc[nt] = __builtin_amdgcn_wmma_f32_16x16x4_f32(
            false, a, false, b, (short)0, acc[nt], false, false);
      }
    }
  }

  // Epilogue: C/D layout -> VGPR g holds (M = g or g+8, N = lane&15).
#pragma unroll
  for (int nt = 0; nt < 8; ++nt) {
#pragma unroll
    for (int g = 0; g < 8; ++g) {
      const int row = m0 + ((lane < 16) ? g : (g + 8));
      const int col = nt * 16 + (lane & 15);
      if (row < nrows) {
        float v = acc[nt][g] + bias[col];
        hout[(size_t)row * DF + col] = fmaxf(v, 0.0f);   // ReLU (all 3 layers)
      }
    }
  }
}

// ---------------------------------------------------------------- pooling
__global__ void k_pool(const float* __restrict__ h, const int* __restrict__ batch,
                       float* __restrict__ gsum, float* __restrict__ gcnt, int n) {
  int node = blockIdx.x;
  if (node >= n) return;
  int b = batch[node];
  int f = threadIdx.x;  // 128 threads
  atomicAdd(&gsum[b * DF + f], h[(size_t)node * DF + f]);
  if (f == 0) atomicAdd(&gcnt[b], 1.0f);
}

__global__ void k_final(const float* __restrict__ gsum, const float* __restrict__ gcnt,
                        const float* __restrict__ wlin, const float* __restrict__ blin,
                        float* __restrict__ out) {
  int idx = blockIdx.x * blockDim.x + threadIdx.x;
  if (idx >= NGRAPH * DOUT) return;
  int g = idx / DOUT;
  int o = idx - g * DOUT;
  float inv = 1.0f / fmaxf(gcnt[g], 1.0f);
  float s = 0.0f;
  for (int k = 0; k < DF; ++k) s += gsum[g * DF + k] * wlin[o * DF + k];
  out[idx] = s * inv + blin[o];
}

// ---------------------------------------------------------------- launch
extern "C" void kernel_launch(void* const* d_in, const int* in_sizes, int n_in,
                              void* d_out, int out_size, void* d_ws, size_t ws_size,
                              hipStream_t stream) {
  (void)in_sizes; (void)n_in; (void)out_size; (void)ws_size;

  const float* x     = (const float*)d_in[0];
  const int*   ei    = (const int*)d_in[1];    // [2][E]
  const int*   batch = (const int*)d_in[2];
  const float* w1l = (const float*)d_in[3];
  const float* b1  = (const float*)d_in[4];
  const float* w1r = (const float*)d_in[5];
  const float* w2l = (const float*)d_in[6];
  const float* b2  = (const float*)d_in[7];
  const float* w2r = (const float*)d_in[8];
  const float* w3l = (const float*)d_in[9];
  const float* b3  = (const float*)d_in[10];
  const float* w3r = (const float*)d_in[11];
  const float* wlin = (const float*)d_in[12];
  const float* blin = (const float*)d_in[13];
  float* out = (float*)d_out;

  const int* src = ei;            // edge_index[0]
  const int* dst = ei + NEDGES;   // edge_index[1]

  // Workspace layout (floats): ~77 MB total.
  float* ws   = (float*)d_ws;
  float* agg  = ws;                                  // N*128 scatter accumulator
  float* bufB = agg  + (size_t)NNODES * DF;          // N*128 activations ping
  float* bufC = bufB + (size_t)NNODES * DF;          // N*128 activations pong
  float* ideg = bufC + (size_t)NNODES * DF;          // N     1/max(deg,1)
  float* gsum = ideg + NNODES;                       // 64*128 pooled sums
  float* gcnt = gsum + NGRAPH * DF;                  // 64     pooled counts

  dim3 b256(256);

  // Degrees are identical for all 3 layers: compute once.
  k_zero<<<256, b256, 0, stream>>>(ideg, NNODES);
  k_degree<<<(NEDGES + 255) / 256, b256, 0, stream>>>(dst, ideg, NEDGES);
  k_inv_degree<<<(NNODES + 255) / 256, b256, 0, stream>>>(ideg, NNODES);

  const int gemm_grid = (NNODES + 127) / 128;  // 391
  const int scat_grid = (NEDGES + 7) / 8;      // one wave per edge, 8 waves/block

  struct Layer { const float* hin; float* hout; const float* wl; const float* wr; const float* bias; };
  Layer layers[3] = {
    { x,    bufB, w1l, w1r, b1 },
    { bufB, bufC, w2l, w2r, b2 },
    { bufC, bufB, w3l, w3r, b3 },
  };

  for (int l = 0; l < 3; ++l) {
    k_zero<<<1024, b256, 0, stream>>>(agg, (long long)NNODES * DF);
    k_scatter<<<scat_grid, b256, 0, stream>>>(layers[l].hin, src, dst, agg, NEDGES);
    k_sage_gemm<<<gemm_grid, b256, 0, stream>>>(agg, ideg, layers[l].hin,
        layers[l].wl, layers[l].wr, layers[l].bias, layers[l].hout, NNODES);
  }

  // Global mean pool + final linear.
  k_zero<<<32, b256, 0, stream>>>(gsum, (long long)(NGRAPH * DF + NGRAPH));
  k_pool<<<NNODES, dim3(DF), 0, stream>>>(bufB, batch, gsum, gcnt, NNODES);
  k_final<<<(NGRAPH * DOUT + 255) / 256, b256, 0, stream>>>(gsum, gcnt, wlin, blin, out);
}